// fMRI3DGNN_16595753631829
// MI455X (gfx1250) — compile-verified
//
#include <hip/hip_runtime.h>
#include <math.h>

typedef float v2f __attribute__((ext_vector_type(2)));
typedef float v4f __attribute__((ext_vector_type(4)));
typedef float v8f __attribute__((ext_vector_type(8)));

#define N_NODES 200
#define ADJ_N   40000            // 200*200
#define KS      16               // K-split for stage-1 GEMM
#define KC      2500             // 40000 / KS
#define TKC     64               // K-chunk for LDS-staged WMMA GEMM

__device__ __forceinline__ float gelu_exact(float x) {
  return 0.5f * x * (1.0f + erff(x * 0.70710678118654752f));
}

// ---------------------------------------------------------------------------
// Stage 1: partial[chunk][b][n] = sum_{k in chunk} raw_fc[b,k] * W_gb[k,n]
// Memory-bound rank-4 update: stream W_gb rows coalesced (float4/lane),
// raw_fc slice staged in LDS. 40 column tiles x 16 K-chunks = 640 blocks.
// ---------------------------------------------------------------------------
__global__ __launch_bounds__(256) void adj_gemm_partial(
    const float* __restrict__ rawfc,   // 4 x 40000
    const float* __restrict__ Wgb,     // 40000 x 40000
    float* __restrict__ partial)       // KS x 4 x 40000
{
  __shared__ __align__(16) float aS[KC * 4];   // [k][b], 40 KB
  const int chunk = blockIdx.y;
  const int k0 = chunk * KC;
  for (int i = threadIdx.x; i < KC * 4; i += blockDim.x) {
    int kk = i >> 2, b = i & 3;
    aS[i] = rawfc[b * ADJ_N + k0 + kk];
  }
  __syncthreads();

  const int col0 = blockIdx.x * 1024 + threadIdx.x * 4;
  if (col0 >= ADJ_N) return;   // ADJ_N % 4 == 0 so full float4 in range

  float acc[4][4] = {};
  for (int kk = 0; kk < KC; ++kk) {
    if ((kk & 7) == 0 && kk + 64 < KC)
      __builtin_prefetch(Wgb + (size_t)(k0 + kk + 64) * ADJ_N + col0, 0, 1);
    v4f w = *(const v4f*)(Wgb + (size_t)(k0 + kk) * ADJ_N + col0);
    v4f a = *(const v4f*)(aS + kk * 4);
#pragma unroll
    for (int b = 0; b < 4; ++b) {
      acc[b][0] += a[b] * w[0];
      acc[b][1] += a[b] * w[1];
      acc[b][2] += a[b] * w[2];
      acc[b][3] += a[b] * w[3];
    }
  }
#pragma unroll
  for (int b = 0; b < 4; ++b) {
    v4f o = { acc[b][0], acc[b][1], acc[b][2], acc[b][3] };
    *(v4f*)(partial + (size_t)(chunk * 4 + b) * ADJ_N + col0) = o;
  }
}

// Reduce K-splits, add bias, sigmoid.
__global__ __launch_bounds__(256) void adj_reduce_sigmoid(
    const float* __restrict__ partial, const float* __restrict__ bgb,
    float* __restrict__ adjraw)
{
  int idx = blockIdx.x * blockDim.x + threadIdx.x;
  if (idx >= 4 * ADJ_N) return;
  int b = idx / ADJ_N, n = idx - b * ADJ_N;
  float s = bgb[n];
#pragma unroll
  for (int c = 0; c < KS; ++c) s += partial[(size_t)(c * 4 + b) * ADJ_N + n];
  adjraw[idx] = 1.0f / (1.0f + expf(-s));
}

__global__ __launch_bounds__(256) void adj_symmetrize(
    const float* __restrict__ adjraw, float* __restrict__ adjsym)
{
  int idx = blockIdx.x * blockDim.x + threadIdx.x;
  if (idx >= 4 * ADJ_N) return;
  int b = idx / ADJ_N, n = idx - b * ADJ_N;
  int s = n / N_NODES, t = n - s * N_NODES;
  float a = adjraw[b * ADJ_N + s * N_NODES + t];
  float bb = adjraw[b * ADJ_N + t * N_NODES + s];
  adjsym[idx] = 0.5f * (a + bb);
}

// ---------------------------------------------------------------------------
// Exact k-th smallest via 4-pass MSD radix select (positive floats => bit
// pattern is order-preserving). One block per (batch, rank-of-two).
// ---------------------------------------------------------------------------
__global__ __launch_bounds__(256) void quantile_select(
    const float* __restrict__ adjsym, float* __restrict__ kth)
{
  int b = blockIdx.x >> 1;
  int which = blockIdx.x & 1;
  unsigned k = 29999u + (unsigned)which;     // 0.75*(40000-1) = 29999.25
  const float* data = adjsym + (size_t)b * ADJ_N;

  __shared__ unsigned hist[256];
  __shared__ unsigned s_prefix, s_rank;
  if (threadIdx.x == 0) { s_prefix = 0u; s_rank = k; }
  __syncthreads();

  for (int pass = 0; pass < 4; ++pass) {
    int shift = 24 - pass * 8;
    if (threadIdx.x < 256) hist[threadIdx.x] = 0u;
    __syncthreads();
    unsigned prefix = s_prefix;
    unsigned pmask = pass ? (0xFFFFFFFFu << (shift + 8)) : 0u;
    for (int i = threadIdx.x; i < ADJ_N; i += blockDim.x) {
      unsigned u = __float_as_uint(data[i]);
      if ((u & pmask) == prefix) atomicAdd(&hist[(u >> shift) & 0xFF], 1u);
    }
    __syncthreads();
    if (threadIdx.x == 0) {
      unsigned r = s_rank, cum = 0u; int bin = 0;
      for (; bin < 256; ++bin) {
        unsigned c = hist[bin];
        if (cum + c > r) break;
        cum += c;
      }
      s_rank = r - cum;
      s_prefix = prefix | ((unsigned)bin << shift);
    }
    __syncthreads();
  }
  if (threadIdx.x == 0) kth[b * 2 + which] = __uint_as_float(s_prefix);
}

__global__ void quantile_interp(const float* __restrict__ kth,
                                float* __restrict__ thr)
{
  int b = threadIdx.x;
  if (b < 4) {
    float lo = kth[b * 2], hi = kth[b * 2 + 1];
    thr[b] = lo + 0.25f * (hi - lo);   // linear interpolation at 29999.25
  }
}

// ---------------------------------------------------------------------------
// Row stats (mean, std ddof=1) -> 2->8->16 MLP with exact GELU -> LayerNorm.
// One thread per (batch, node).
// ---------------------------------------------------------------------------
__global__ __launch_bounds__(256) void stats_mlp_ln(
    const float* __restrict__ adjsym,
    const float* __restrict__ Wfe1, const float* __restrict__ bfe1,
    const float* __restrict__ Wfe2, const float* __restrict__ bfe2,
    const float* __restrict__ lng,  const float* __restrict__ lnb,
    float* __restrict__ hout)       // 4 x 200 x 16
{
  int idx = blockIdx.x * blockDim.x + threadIdx.x;
  if (idx >= 4 * N_NODES) return;
  int b = idx / N_NODES, s = idx - b * N_NODES;
  const float* row = adjsym + (size_t)b * ADJ_N + s * N_NODES;
  float sum = 0.f, sq = 0.f;
  for (int t = 0; t < N_NODES; ++t) { float v = row[t]; sum += v; sq += v * v; }
  float mean = sum / 200.0f;
  float var1 = (sq - 200.0f * mean * mean) / 199.0f;   // ddof=1
  float sd = sqrtf(fmaxf(var1, 0.f));

  float h1[8];
#pragma unroll
  for (int j = 0; j < 8; ++j)
    h1[j] = gelu_exact(mean * Wfe1[j] + sd * Wfe1[8 + j] + bfe1[j]);

  float h2[16];
#pragma unroll
  for (int i = 0; i < 16; ++i) {
    float z = bfe2[i];
#pragma unroll
    for (int j = 0; j < 8; ++j) z += h1[j] * Wfe2[j * 16 + i];
    h2[i] = z;
  }
  float mu = 0.f;
#pragma unroll
  for (int i = 0; i < 16; ++i) mu += h2[i];
  mu *= (1.0f / 16.0f);
  float var = 0.f;
#pragma unroll
  for (int i = 0; i < 16; ++i) { float d = h2[i] - mu; var += d * d; }
  var *= (1.0f / 16.0f);
  float inv = rsqrtf(var + 1e-5f);
#pragma unroll
  for (int i = 0; i < 16; ++i)
    hout[idx * 16 + i] = (h2[i] - mu) * inv * lng[i] + lnb[i];
}

// ---------------------------------------------------------------------------
// LDS-staged f32 WMMA GEMM for K % 64 == 0, N % 128 == 0, M % 16 == 0.
// Block = 8 waves computes a 16(M) x 128(N) band; A tile (16xTKC) shared by
// all 8 waves via LDS, B chunk (TKC x 128) staged coalesced to LDS once.
// Per chunk each wave runs 16 V_WMMA_F32_16X16X4_F32 fed from ds_load.
// Fragment layouts:
//  A-frag: lane L holds A[(L&15)][kbase + {0,1}], kbase = (L>>4)*2.
//  B-frag: lane L holds B[kbase + {0,1}][wave*16 + (L&15)].
//  C/D:    VGPR v holds C[mt*16 + (L>>4)*8 + v][col].
// ---------------------------------------------------------------------------
__global__ __launch_bounds__(256) void wmma_gemm_bias_lds(
    const float* __restrict__ A, const float* __restrict__ B,
    const float* __restrict__ bias, float* __restrict__ C,
    int M, int K, int N)
{
  __shared__ __align__(16) float As[16 * TKC];    // 4 KB
  __shared__ __align__(16) float Bs[TKC * 128];   // 32 KB
  const int mt  = blockIdx.y;          // M tile (16 rows)
  const int nb  = blockIdx.x;          // N band (128 cols)
  const int tid = threadIdx.x;
  const int wave = tid >> 5, lane = tid & 31;
  const int col = nb * 128 + wave * 16 + (lane & 15);
  const int kbase = (lane >> 4) * 2;   // 0 or 2

  v8f acc = {};
  for (int k0 = 0; k0 < K; k0 += TKC) {
    // Stage A: 16 x TKC = 1024 floats, one v4f per thread, coalesced rows.
    {
      int i = tid * 4;                       // 0..1023
      int r = i >> 6, c = i & (TKC - 1);     // TKC == 64
      *(v4f*)(As + i) = *(const v4f*)(A + (size_t)(mt * 16 + r) * K + k0 + c);
    }
    // Stage B: TKC x 128 = 8192 floats, 8 x (256 threads x v4f), coalesced.
#pragma unroll
    for (int it = 0; it < 8; ++it) {
      int i = (it * 256 + tid) * 4;          // 0..8191
      int r = i >> 7, c = i & 127;
      *(v4f*)(Bs + i) = *(const v4f*)(B + (size_t)(k0 + r) * N + nb * 128 + c);
    }
    __syncthreads();

    const float* ap = As + (lane & 15) * TKC + kbase;
    const float* bp = Bs + (size_t)kbase * 128 + wave * 16 + (lane & 15);
#pragma unroll
    for (int kk = 0; kk < TKC; kk += 4) {
      v2f a, bfrag;
      a[0] = ap[kk];          a[1] = ap[kk + 1];
      bfrag[0] = bp[kk * 128]; bfrag[1] = bp[(kk + 1) * 128];
      acc = __builtin_amdgcn_wmma_f32_16x16x4_f32(
          false, a, false, bfrag, (short)0, acc, false, false);
    }
    __syncthreads();
  }

  int rbase = mt * 16 + (lane >> 4) * 8;
  float bv = bias[col];
#pragma unroll
  for (int v = 0; v < 8; ++v)
    C[(size_t)(rbase + v) * N + col] = acc[v] + bv;
}

// Direct-load WMMA GEMM for small K (layer 1, K=16): one 16x16 tile per wave.
__global__ __launch_bounds__(256) void wmma_gemm_bias_small(
    const float* __restrict__ A, const float* __restrict__ B,
    const float* __restrict__ bias, float* __restrict__ C,
    int M, int K, int N)
{
  int wid  = (blockIdx.x * blockDim.x + threadIdx.x) >> 5;
  int lane = threadIdx.x & 31;
  int ntiles = N >> 4;
  int mt = wid / ntiles, nt = wid - mt * ntiles;
  if (mt * 16 >= M) return;                       // wave-uniform exit

  int row = mt * 16 + (lane & 15);
  int col = nt * 16 + (lane & 15);
  int kbase = (lane >> 4) * 2;

  v8f acc = {};
  const float* ap = A + (size_t)row * K + kbase;
  const float* bp = B + (size_t)kbase * N + col;
  for (int k = 0; k < K; k += 4) {
    v2f a, bfrag;
    a[0] = ap[k];                 a[1] = ap[k + 1];
    bfrag[0] = bp[(size_t)k * N]; bfrag[1] = bp[(size_t)(k + 1) * N];
    acc = __builtin_amdgcn_wmma_f32_16x16x4_f32(
        false, a, false, bfrag, (short)0, acc, false, false);
  }

  int rbase = mt * 16 + (lane >> 4) * 8;
  float bv = bias[col];
#pragma unroll
  for (int v = 0; v < 8; ++v)
    C[(size_t)(rbase + v) * N + col] = acc[v] + bv;
}

// ---------------------------------------------------------------------------
// GATv2 attention + aggregation + GELU, one block per (batch, head, target).
// ---------------------------------------------------------------------------
__global__ __launch_bounds__(256) void gat_attention(
    const float* __restrict__ xl,    // (B*N, H*C)
    const float* __restrict__ xr,    // (B*N, H*C)
    const float* __restrict__ att,   // (H, C)
    const float* __restrict__ bo,    // (H*C)
    const float* __restrict__ adjsym,// (B, N, N)
    const float* __restrict__ thr,   // (B)
    float* __restrict__ xout,        // (B*N, H*C)
    int H, int C)
{
  __shared__ float xrS[512];
  __shared__ float attS[512];
  __shared__ float logitsS[208];
  __shared__ float redS[256];

  int bid = blockIdx.x;
  int t  = bid % N_NODES;
  int bh = bid / N_NODES;
  int h0 = bh % H;
  int b  = bh / H;
  int tid = threadIdx.x;
  int wave = tid >> 5, lane = tid & 31;
  size_t HC = (size_t)H * C;

  for (int c = tid; c < C; c += 256) {
    xrS[c]  = xr[(size_t)(b * N_NODES + t) * HC + (size_t)h0 * C + c];
    attS[c] = att[h0 * C + c];
  }
  __syncthreads();

  // logits[s] = sum_c leakyrelu(xl[s,c] + xr[t,c], 0.2) * att[c]
  for (int s = wave; s < N_NODES; s += 8) {
    const float* xls = xl + (size_t)(b * N_NODES + s) * HC + (size_t)h0 * C;
    float acc = 0.f;
    for (int c = lane; c < C; c += 32) {
      float v = xls[c] + xrS[c];
      v = (v > 0.f) ? v : 0.2f * v;
      acc += v * attS[c];
    }
#pragma unroll
    for (int off = 16; off > 0; off >>= 1) acc += __shfl_down(acc, off, 32);
    if (lane == 0) logitsS[s] = acc;
  }
  __syncthreads();

  // masked softmax over senders s
  float thrv = thr[b];
  bool mk = false;
  float lv = -3.402823466e38f;
  if (tid < N_NODES) {
    mk = adjsym[(size_t)b * ADJ_N + (size_t)tid * N_NODES + t] > thrv;
    if (mk) lv = logitsS[tid];
  }
  redS[tid] = lv;
  __syncthreads();
#pragma unroll
  for (int off = 128; off > 0; off >>= 1) {
    if (tid < off) redS[tid] = fmaxf(redS[tid], redS[tid + off]);
    __syncthreads();
  }
  float maxv = redS[0];
  __syncthreads();
  float wv = mk ? expf(logitsS[tid] - maxv) : 0.f;
  redS[tid] = (tid < N_NODES) ? wv : 0.f;
  __syncthreads();
#pragma unroll
  for (int off = 128; off > 0; off >>= 1) {
    if (tid < off) redS[tid] += redS[tid + off];
    __syncthreads();
  }
  float denom = fmaxf(redS[0], 1e-16f);
  __syncthreads();
  if (tid < N_NODES) logitsS[tid] = wv;
  __syncthreads();

  // out[t,c] = gelu( sum_s w[s]*xl[s,c] / denom + bo[c] )
  float inv = 1.0f / denom;
  for (int c = tid; c < C; c += 256) {
    float acc = 0.f;
    for (int s = 0; s < N_NODES; ++s)
      acc += logitsS[s] * xl[(size_t)(b * N_NODES + s) * HC + (size_t)h0 * C + c];
    float o = acc * inv + bo[h0 * C + c];
    xout[(size_t)(b * N_NODES + t) * HC + (size_t)h0 * C + c] = gelu_exact(o);
  }
}

__global__ __launch_bounds__(256) void mean_nodes(
    const float* __restrict__ x, float* __restrict__ out)
{
  int idx = blockIdx.x * blockDim.x + threadIdx.x;
  if (idx >= 4 * 512) return;
  int b = idx / 512, c = idx - b * 512;
  float s = 0.f;
  for (int t = 0; t < N_NODES; ++t)
    s += x[(size_t)(b * N_NODES + t) * 512 + c];
  out[idx] = s * (1.0f / 200.0f);
}

// ---------------------------------------------------------------------------
extern "C" void kernel_launch(void* const* d_in, const int* in_sizes, int n_in,
                              void* d_out, int out_size, void* d_ws, size_t ws_size,
                              hipStream_t stream) {
  (void)in_sizes; (void)n_in; (void)out_size; (void)ws_size;
  const float* raw_fc = (const float*)d_in[0];
  const float* W_gb   = (const float*)d_in[1];
  const float* b_gb   = (const float*)d_in[2];
  const float* W_fe1  = (const float*)d_in[3];
  const float* b_fe1  = (const float*)d_in[4];
  const float* W_fe2  = (const float*)d_in[5];
  const float* b_fe2  = (const float*)d_in[6];
  const float* ln_g   = (const float*)d_in[7];
  const float* ln_b   = (const float*)d_in[8];
  const float* Wl[3]  = {(const float*)d_in[9],  (const float*)d_in[15], (const float*)d_in[21]};
  const float* bl[3]  = {(const float*)d_in[10], (const float*)d_in[16], (const float*)d_in[22]};
  const float* Wr[3]  = {(const float*)d_in[11], (const float*)d_in[17], (const float*)d_in[23]};
  const float* br[3]  = {(const float*)d_in[12], (const float*)d_in[18], (const float*)d_in[24]};
  const float* att[3] = {(const float*)d_in[13], (const float*)d_in[19], (const float*)d_in[25]};
  const float* bo[3]  = {(const float*)d_in[14], (const float*)d_in[20], (const float*)d_in[26]};

  float* ws      = (float*)d_ws;
  float* partial = ws;                      // 16*4*40000 = 2,560,000
  float* adjraw  = partial + 2560000;       // 160,000
  float* adjsym  = adjraw  + 160000;        // 160,000
  float* kth     = adjsym  + 160000;        // 8 (+pad)
  float* thr     = kth     + 16;            // 4 (+pad)
  float* h0      = thr     + 16;            // 4*200*16 = 12,800
  float* xl      = h0      + 12800;         // 4*200*1024 = 819,200
  float* xr      = xl      + 819200;        // 819,200
  float* x1      = xr      + 819200;        // 819,200
  float* x2      = x1      + 819200;        // 819,200
  float* x3      = x2      + 819200;        // 4*200*512 = 409,600

  // Stage 1: adjacency (HBM-bound: 6.4 GB of W_gb streamed once)
  adj_gemm_partial<<<dim3(40, KS), 256, 0, stream>>>(raw_fc, W_gb, partial);
  adj_reduce_sigmoid<<<(4 * ADJ_N + 255) / 256, 256, 0, stream>>>(partial, b_gb, adjraw);
  adj_symmetrize<<<(4 * ADJ_N + 255) / 256, 256, 0, stream>>>(adjraw, adjsym);

  // Quantile threshold (exact order statistics via radix select)
  quantile_select<<<8, 256, 0, stream>>>(adjsym, kth);
  quantile_interp<<<1, 32, 0, stream>>>(kth, thr);

  // Node features
  stats_mlp_ln<<<(4 * N_NODES + 255) / 256, 256, 0, stream>>>(
      adjsym, W_fe1, b_fe1, W_fe2, b_fe2, ln_g, ln_b, h0);

  const int M = 4 * N_NODES;  // 800
  // Layer 1: fin=16, H=8, C=128 (HC=1024) -- K too small for LDS staging
  wmma_gemm_bias_small<<<(50 * 64 + 7) / 8, 256, 0, stream>>>(h0, Wl[0], bl[0], xl, M, 16, 1024);
  wmma_gemm_bias_small<<<(50 * 64 + 7) / 8, 256, 0, stream>>>(h0, Wr[0], br[0], xr, M, 16, 1024);
  gat_attention<<<4 * 8 * N_NODES, 256, 0, stream>>>(xl, xr, att[0], bo[0], adjsym, thr, x1, 8, 128);

  // Layer 2: fin=1024, H=4, C=256 (HC=1024)
  wmma_gemm_bias_lds<<<dim3(1024 / 128, 50), 256, 0, stream>>>(x1, Wl[1], bl[1], xl, M, 1024, 1024);
  wmma_gemm_bias_lds<<<dim3(1024 / 128, 50), 256, 0, stream>>>(x1, Wr[1], br[1], xr, M, 1024, 1024);
  gat_attention<<<4 * 4 * N_NODES, 256, 0, stream>>>(xl, xr, att[1], bo[1], adjsym, thr, x2, 4, 256);

  // Layer 3: fin=1024, H=1, C=512 (HC=512)
  wmma_gemm_bias_lds<<<dim3(512 / 128, 50), 256, 0, stream>>>(x2, Wl[2], bl[2], xl, M, 1024, 512);
  wmma_gemm_bias_lds<<<dim3(512 / 128, 50), 256, 0, stream>>>(x2, Wr[2], br[2], xr, M, 1024, 512);
  gat_attention<<<4 * 1 * N_NODES, 256, 0, stream>>>(xl, xr, att[2], bo[2], adjsym, thr, x3, 1, 512);

  // Final node mean -> (4, 512)
  mean_nodes<<<(4 * 512 + 255) / 256, 256, 0, stream>>>(x3, (float*)d_out);
}